// HierarchicalSoftmax_67585605370260
// MI455X (gfx1250) — compile-verified
//
#include <hip/hip_runtime.h>

typedef __attribute__((ext_vector_type(2))) float v2f;
typedef __attribute__((ext_vector_type(8))) float v8f;

#define NTOK 4096      // B*L
#define DIM  512       // D
#define NCLS 225       // N_CLASSES
#define NPC  224       // N_PER_CLASS
#define XPAD 516       // LDS row stride for x tiles (bank-conflict-free, 16B-aligned)

__device__ __forceinline__ float wave_max(float v) {
    #pragma unroll
    for (int m = 16; m > 0; m >>= 1) v = fmaxf(v, __shfl_xor(v, m, 32));
    return v;
}
__device__ __forceinline__ float wave_sum(float v) {
    #pragma unroll
    for (int m = 16; m > 0; m >>= 1) v += __shfl_xor(v, m, 32);
    return v;
}

// ---------------------------------------------------------------------------
// Stage A: logits1 = x @ W1 + b1   [NTOK, NCLS]  via V_WMMA_F32_16X16X4_F32
// grid: NTOK/16 blocks, 480 threads (15 waves); wave w owns cols [16w,16w+16)
// Column index is CLAMPED for loads (padding lanes redundantly read col 224)
// so the K-loop has no EXEC-mask manipulation; only the store is guarded.
// ---------------------------------------------------------------------------
__global__ __launch_bounds__(480)
void hsm_class_logits_wmma(const float* __restrict__ x,
                           const float* __restrict__ W1,
                           const float* __restrict__ b1,
                           float* __restrict__ logits)
{
    const int wave  = threadIdx.x >> 5;
    const int lane  = threadIdx.x & 31;
    const int l15   = lane & 15;
    const int lhalf = lane >> 4;
    const int mbase = blockIdx.x * 16;
    const int col   = wave * 16 + l15;              // 0..239
    const bool cok  = (col < NCLS);
    const int colc  = cok ? col : (NCLS - 1);       // clamped for loads

    const float* __restrict__ xrow = x  + (size_t)(mbase + l15) * DIM;
    const float* __restrict__ wcol = W1 + colc;     // + k*NCLS per row

    v8f acc = {};
    for (int k0 = 0; k0 < DIM; k0 += 4) {
        const int ka = k0 + 2 * lhalf;
        v2f a, b;
        a.x = xrow[ka + 0];
        a.y = xrow[ka + 1];
        b.x = wcol[(size_t)(ka + 0) * NCLS];
        b.y = wcol[(size_t)(ka + 1) * NCLS];
        acc = __builtin_amdgcn_wmma_f32_16x16x4_f32(false, a, false, b,
                                                    (short)0, acc, false, false);
    }

    if (cok) {
        const float bias = b1[col];
        #pragma unroll
        for (int r = 0; r < 8; ++r) {
            const int row = mbase + r + 8 * lhalf;
            logits[(size_t)row * NCLS + col] = acc[r] + bias;
        }
    }
}

// ---------------------------------------------------------------------------
// class_p[n] = softmax(logits[n,:])[cls[n]] ; one wave per token
// ---------------------------------------------------------------------------
__global__ __launch_bounds__(256)
void hsm_class_p(const float* __restrict__ logits,
                 const int*   __restrict__ target,
                 float*       __restrict__ classp)
{
    const int n    = blockIdx.x * 8 + (threadIdx.x >> 5);
    const int lane = threadIdx.x & 31;
    if (n >= NTOK) return;

    const float* __restrict__ row = logits + (size_t)n * NCLS;
    float m = -3.0e38f;
    for (int t = lane; t < NCLS; t += 32) m = fmaxf(m, row[t]);
    m = wave_max(m);
    float s = 0.0f;
    for (int t = lane; t < NCLS; t += 32) s += __expf(row[t] - m);
    s = wave_sum(s);
    if (lane == 0) {
        const int c = target[n] / NPC;
        classp[n] = __expf(row[c] - m) / s;
    }
}

// ---------------------------------------------------------------------------
// Bucketing: histogram -> exclusive scan -> scatter (counting sort by class)
// ---------------------------------------------------------------------------
__global__ void hsm_bucket_zero(int* __restrict__ counts)
{
    const int t = blockIdx.x * blockDim.x + threadIdx.x;
    if (t < NCLS) counts[t] = 0;
}

__global__ void hsm_hist(const int* __restrict__ target, int* __restrict__ counts)
{
    const int n = blockIdx.x * blockDim.x + threadIdx.x;
    if (n < NTOK) atomicAdd(&counts[target[n] / NPC], 1);
}

__global__ __launch_bounds__(256)
void hsm_scan(const int* __restrict__ counts,
              int* __restrict__ starts, int* __restrict__ cursor)
{
    __shared__ int sc[256];
    const int t = threadIdx.x;
    const int own = (t < NCLS) ? counts[t] : 0;
    sc[t] = own;
    __syncthreads();
    for (int off = 1; off < 256; off <<= 1) {
        const int v = (t >= off) ? sc[t - off] : 0;
        __syncthreads();
        sc[t] += v;
        __syncthreads();
    }
    if (t < NCLS) {
        const int excl = sc[t] - own;
        starts[t] = excl;
        cursor[t] = excl;
    }
}

__global__ void hsm_scatter(const int* __restrict__ target,
                            int* __restrict__ cursor, int* __restrict__ tokenIdx)
{
    const int n = blockIdx.x * blockDim.x + threadIdx.x;
    if (n < NTOK) {
        const int c = target[n] / NPC;
        const int pos = atomicAdd(&cursor[c], 1);
        tokenIdx[pos] = n;
    }
}

// ---------------------------------------------------------------------------
// Per-class WMMA GEMM + inner softmax + output.
// grid: NCLS blocks, 448 threads (14 waves x 16 cols = 224 exactly).
// Chunks of 32 tokens: one B-fragment feeds two WMMAs (rows 0-15 / 16-31).
// x rows staged to LDS as float4 (global_load_b128 / ds_store_b128).
// ---------------------------------------------------------------------------
__global__ __launch_bounds__(448)
void hsm_inner_class_wmma(const float* __restrict__ x,
                          const int*   __restrict__ target,
                          const int*   __restrict__ starts,
                          const int*   __restrict__ counts,
                          const int*   __restrict__ tokenIdx,
                          const float* __restrict__ W2,
                          const float* __restrict__ b2,
                          const float* __restrict__ classp,
                          float*       __restrict__ out)
{
    __shared__ __align__(16) float xs[32 * XPAD];   // 32 gathered x rows, padded
    __shared__ float ltile[32][NPC];                // inner logits tile
    __shared__ int   toks[32];

    const int c     = blockIdx.x;
    const int start = starts[c];
    const int cnt   = counts[c];
    if (cnt == 0) return;

    const int tid   = threadIdx.x;
    const int wave  = tid >> 5;
    const int lane  = tid & 31;
    const int l15   = lane & 15;
    const int lhalf = lane >> 4;
    const int col   = wave * 16 + l15;              // 0..223, always valid

    const float bias = b2[(size_t)c * NPC + col];
    const float* __restrict__ w2c = W2 + (size_t)c * DIM * NPC;

    for (int cs = 0; cs < cnt; cs += 32) {
        if (tid < 32) {
            const int r = cs + tid;
            toks[tid] = (r < cnt) ? tokenIdx[start + r] : -1;
        }
        __syncthreads();

        // stage 32 gathered x rows into LDS as float4 (zeros for padding rows)
        for (int i = tid; i < 32 * (DIM / 4); i += 448) {
            const int row = i >> 7;                 // / (DIM/4)
            const int d4  = i & (DIM / 4 - 1);
            const int tok = toks[row];
            float4 v = make_float4(0.f, 0.f, 0.f, 0.f);
            if (tok >= 0)
                v = ((const float4*)(x + (size_t)tok * DIM))[d4];
            *((float4*)(xs + row * XPAD + d4 * 4)) = v;
        }
        __syncthreads();

        v8f acc0 = {}, acc1 = {};
        for (int k0 = 0; k0 < DIM; k0 += 4) {
            const int ka = k0 + 2 * lhalf;
            v2f a0, a1, b;
            a0.x = xs[l15 * XPAD + ka];
            a0.y = xs[l15 * XPAD + ka + 1];
            a1.x = xs[(16 + l15) * XPAD + ka];
            a1.y = xs[(16 + l15) * XPAD + ka + 1];
            b.x  = w2c[(size_t)(ka + 0) * NPC + col];
            b.y  = w2c[(size_t)(ka + 1) * NPC + col];
            acc0 = __builtin_amdgcn_wmma_f32_16x16x4_f32(false, a0, false, b,
                                                         (short)0, acc0, false, false);
            acc1 = __builtin_amdgcn_wmma_f32_16x16x4_f32(false, a1, false, b,
                                                         (short)0, acc1, false, false);
        }

        // D layout: VGPR r -> row r (lanes 0-15) / row r+8 (lanes 16-31)
        #pragma unroll
        for (int r = 0; r < 8; ++r) {
            ltile[r + 8 * lhalf][col]      = acc0[r] + bias;
            ltile[16 + r + 8 * lhalf][col] = acc1[r] + bias;
        }
        __syncthreads();

        // inner softmax: one wave per row (rows wave, wave+14, wave+28)
        for (int m = wave; m < 32; m += 14) {
            const int r = cs + m;
            if (r < cnt) {
                float vals[7];
                float vmax = -3.0e38f;
                #pragma unroll
                for (int j = 0; j < 7; ++j) {
                    vals[j] = ltile[m][lane + 32 * j];   // 224 = 32*7 exact
                    vmax = fmaxf(vmax, vals[j]);
                }
                vmax = wave_max(vmax);
                float ssum = 0.0f;
                #pragma unroll
                for (int j = 0; j < 7; ++j) ssum += __expf(vals[j] - vmax);
                ssum = wave_sum(ssum);
                if (lane == 0) {
                    const int n   = toks[m];
                    const int tgt = target[n];
                    const int idx = tgt - (tgt / NPC) * NPC;
                    const float li = ltile[m][idx];
                    out[n] = classp[n] * (__expf(li - vmax) / ssum);
                }
            }
        }
        __syncthreads();
    }
}

// ---------------------------------------------------------------------------
// Harness entry.
// in order: x[8*512*512] f32, target[8*512] i32, W1[512*225] f32,
//           b1[225] f32, W2[225*512*224] f32, b2[225*224] f32
// out: [8, 512, 1] f32 (4096 floats)
// ---------------------------------------------------------------------------
extern "C" void kernel_launch(void* const* d_in, const int* in_sizes, int n_in,
                              void* d_out, int out_size, void* d_ws, size_t ws_size,
                              hipStream_t stream)
{
    (void)in_sizes; (void)n_in; (void)out_size; (void)ws_size;

    const float* x      = (const float*)d_in[0];
    const int*   target = (const int*)  d_in[1];
    const float* W1     = (const float*)d_in[2];
    const float* b1     = (const float*)d_in[3];
    const float* W2     = (const float*)d_in[4];
    const float* b2     = (const float*)d_in[5];
    float*       out    = (float*)d_out;

    // workspace layout (bytes)
    char* ws = (char*)d_ws;
    float* logits   = (float*)(ws);                         // 4096*225*4 = 3,686,400
    float* classp   = (float*)(ws + 3686400);               // 4096*4     =    16,384
    int*   counts   = (int*)  (ws + 3702784 + 0 * 1024);    // 225*4
    int*   starts   = (int*)  (ws + 3702784 + 1 * 1024);    // 225*4
    int*   cursor   = (int*)  (ws + 3702784 + 2 * 1024);    // 225*4
    int*   tokenIdx = (int*)  (ws + 3702784 + 3 * 1024);    // 4096*4

    hsm_class_logits_wmma<<<NTOK / 16, 480, 0, stream>>>(x, W1, b1, logits);
    hsm_class_p<<<NTOK / 8, 256, 0, stream>>>(logits, target, classp);
    hsm_bucket_zero<<<1, 256, 0, stream>>>(counts);
    hsm_hist<<<NTOK / 256, 256, 0, stream>>>(target, counts);
    hsm_scan<<<1, 256, 0, stream>>>(counts, starts, cursor);
    hsm_scatter<<<NTOK / 256, 256, 0, stream>>>(target, cursor, tokenIdx);
    hsm_inner_class_wmma<<<NCLS, 448, 0, stream>>>(x, target, starts, counts,
                                                   tokenIdx, W2, b2, classp, out);
}